// CINLayer_15960098472080
// MI455X (gfx1250) — compile-verified
//
#include <hip/hip_runtime.h>

// CIN (xDeepFM) fused 3-layer kernel for gfx1250 (MI455X).
// BATCH=1024, F0=32, EMB=64, layer sizes (128,128,128), fp32 throughout.
// Per batch: cur = W^T (128 x fanin) @ Z (fanin x 64), Z[ij,e]=x[i,e]*h[j,e],
// computed with V_WMMA_F32_16X16X4_F32 (exact fp32 matrix path on CDNA5).
//
// Tiling: 8 waves / batch; wave w owns M-tiles {2*(w&3), 2*(w&3)+1} (rows)
// and N-tiles {2*(w>>2), 2*(w>>2)+1} (emb cols). Per K=4 chunk this issues
// 4 WMMAs against only 2 ds_load_2addr (h) + 2 dual-muls (B side shared
// across both M-tiles) and 4 coalesced W loads (A side shared across both
// N-tiles). x factors are hoisted over the inner j-loop (i changes slowly).

typedef float v2f __attribute__((ext_vector_type(2)));
typedef float v8f __attribute__((ext_vector_type(8)));

#define F0    32
#define EMB   64
#define KOUT  128
#define NOUT  256   // outputs per batch row: 64 + 64 + 128

__device__ __forceinline__ v8f wmma_f32(v2f a, v2f b, v8f c) {
    return __builtin_amdgcn_wmma_f32_16x16x4_f32(
        false, a, false, b, (short)0, c, false, false);
}

// One layer: all 8 waves participate (EXEC all-ones for WMMA).
__device__ __forceinline__ void cin_layer_wmma(
    const float* __restrict__ xs,   // LDS: F0 x EMB  (row-major, e fastest)
    const float* __restrict__ h,    // LDS: fi x EMB
    int fi,
    const float* __restrict__ W,    // global: (F0*fi) x KOUT row-major
    const float* __restrict__ bias, // global: KOUT
    float* __restrict__ cur,        // LDS: KOUT x EMB
    int wave, int lane)
{
    const int m     = lane & 15;   // intra-tile index
    const int hi    = lane >> 4;   // half-wave select (K pairs 0/1 vs 2/3)
    const int mpair = wave & 3;    // M-tile pair -> rows [mpair*32, +32)
    const int epair = wave >> 2;   // N-tile pair -> cols [epair*32, +32)

    const int kA = mpair * 32 + m;       // W column for A-tile 0 (A-tile 1: +16)
    const int e0 = epair * 32 + m;       // emb col for N-tile 0
    const int e1 = e0 + 16;              // emb col for N-tile 1

    v8f acc[2][2];
#pragma unroll
    for (int mt = 0; mt < 2; ++mt)
#pragma unroll
        for (int nt = 0; nt < 2; ++nt)
            acc[mt][nt] = (v8f){0,0,0,0,0,0,0,0};

    for (int i0 = 0; i0 < F0; ++i0) {
        // x factors: constant over the whole inner j-loop
        const float xv0 = xs[i0 * EMB + e0];
        const float xv1 = xs[i0 * EMB + e1];
        const float* Wrow = W + (size_t)i0 * fi * KOUT;

        for (int jb = 0; jb < fi; jb += 4) {
            const int j0 = jb + 2 * hi;       // even; j0 and j0+1 share i0
            // ---- A tiles (16x4 of W^T): lanes 0-15 carry K=0,1; 16-31 K=2,3
            const float* wp = Wrow + (size_t)j0 * KOUT;
            v2f a0, a1;
            a0.x = wp[kA];              a0.y = wp[KOUT + kA];
            a1.x = wp[kA + 16];         a1.y = wp[KOUT + kA + 16];

            // ---- B tiles: B[K=2*hi+v, n] = x[i0,e]*h[j0+v,e]
            const float* hp = h + j0 * EMB;
            const float h00 = hp[e0];
            const float h01 = hp[EMB + e0];
            const float h10 = hp[e1];
            const float h11 = hp[EMB + e1];
            v2f bv0, bv1;
            bv0.x = xv0 * h00;  bv0.y = xv0 * h01;
            bv1.x = xv1 * h10;  bv1.y = xv1 * h11;

            acc[0][0] = wmma_f32(a0, bv0, acc[0][0]);
            acc[0][1] = wmma_f32(a0, bv1, acc[0][1]);
            acc[1][0] = wmma_f32(a1, bv0, acc[1][0]);
            acc[1][1] = wmma_f32(a1, bv1, acc[1][1]);
        }
    }

    // ---- store C tiles (+bias) to LDS cur; C layout: VGPR r,
    // lanes 0-15 -> M=r, lanes 16-31 -> M=r+8, N = lane&15
#pragma unroll
    for (int mt = 0; mt < 2; ++mt) {
#pragma unroll
        for (int r = 0; r < 8; ++r) {
            const int row = (2 * mpair + mt) * 16 + r + 8 * hi;
            const float bb = bias[row];
#pragma unroll
            for (int nt = 0; nt < 2; ++nt) {
                cur[row * EMB + epair * 32 + nt * 16 + m] = acc[mt][nt][r] + bb;
            }
        }
    }
}

__global__ __launch_bounds__(256, 1) void cin_fused_kernel(
    const float* __restrict__ x,
    const float* __restrict__ W0, const float* __restrict__ b0,
    const float* __restrict__ W1, const float* __restrict__ b1,
    const float* __restrict__ W2, const float* __restrict__ b2,
    float* __restrict__ out)
{
    __shared__ float xs[F0 * EMB];     //  8 KB
    __shared__ float hs[64 * EMB];     // 16 KB
    __shared__ float cur[KOUT * EMB];  // 32 KB

    const int b    = blockIdx.x;
    const int tid  = threadIdx.x;
    const int wave = tid >> 5;
    const int lane = tid & 31;

    // stage x[b] into LDS (coalesced)
    const float* xb = x + (size_t)b * F0 * EMB;
    for (int idx = tid; idx < F0 * EMB; idx += 256) xs[idx] = xb[idx];
    __syncthreads();

    // ---------------- layer 0: hidden = x (fi = 32) ----------------
    cin_layer_wmma(xs, xs, 32, W0, b0, cur, wave, lane);
    __syncthreads();
    for (int idx = tid; idx < 64 * EMB; idx += 256) hs[idx] = cur[idx];
    if (tid < 64) {   // direct part: rows 64..127, sum over e
        float s = 0.f;
#pragma unroll
        for (int e = 0; e < EMB; ++e)
            s += cur[(64 + tid) * EMB + ((e + tid) & (EMB - 1))]; // bank-rotated
        out[(size_t)b * NOUT + tid] = s;
    }
    __syncthreads();

    // ---------------- layer 1 (fi = 64) ----------------
    cin_layer_wmma(xs, hs, 64, W1, b1, cur, wave, lane);
    __syncthreads();
    for (int idx = tid; idx < 64 * EMB; idx += 256) hs[idx] = cur[idx];
    if (tid < 64) {
        float s = 0.f;
#pragma unroll
        for (int e = 0; e < EMB; ++e)
            s += cur[(64 + tid) * EMB + ((e + tid) & (EMB - 1))];
        out[(size_t)b * NOUT + 64 + tid] = s;
    }
    __syncthreads();

    // ---------------- layer 2 (fi = 64, all 128 rows direct) ----------------
    cin_layer_wmma(xs, hs, 64, W2, b2, cur, wave, lane);
    __syncthreads();
    if (tid < 128) {
        float s = 0.f;
#pragma unroll
        for (int e = 0; e < EMB; ++e)
            s += cur[tid * EMB + ((e + tid) & (EMB - 1))];
        out[(size_t)b * NOUT + 128 + tid] = s;
    }
}

extern "C" void kernel_launch(void* const* d_in, const int* in_sizes, int n_in,
                              void* d_out, int out_size, void* d_ws, size_t ws_size,
                              hipStream_t stream) {
    (void)in_sizes; (void)n_in; (void)out_size; (void)d_ws; (void)ws_size;
    const float* x  = (const float*)d_in[0];
    const float* W0 = (const float*)d_in[1];
    const float* b0 = (const float*)d_in[2];
    const float* W1 = (const float*)d_in[3];
    const float* b1 = (const float*)d_in[4];
    const float* W2 = (const float*)d_in[5];
    const float* b2 = (const float*)d_in[6];
    float* out = (float*)d_out;
    cin_fused_kernel<<<dim3(1024), dim3(256), 0, stream>>>(
        x, W0, b0, W1, b1, W2, b2, out);
}